// CandidateFinder_14474039787699
// MI455X (gfx1250) — compile-verified
//
#include <hip/hip_runtime.h>
#include <stdint.h>

#ifndef USE_ASYNC_LDS
#define USE_ASYNC_LDS 1
#endif

#define BB      4
#define LL      4096
#define DD      8
#define KMAX    64
#define NCODES  256

// ---------------------------------------------------------------------------
// Kernel 1: pack sign bits of 8 floats into one 8-bit code per row.
// 2*B*L = 32768 threads, two global_load_b128 per thread. Bandwidth-trivial.
// ---------------------------------------------------------------------------
__global__ void CandidateFinder_pack_codes(const float* __restrict__ q,
                                           const float* __restrict__ k,
                                           int* __restrict__ qcodes,
                                           int* __restrict__ kcodes) {
    const int N = BB * LL;
    int t = blockIdx.x * blockDim.x + threadIdx.x;
    if (t >= 2 * N) return;
    const float* src;
    int* dst;
    int i;
    if (t < N) { src = q; dst = qcodes; i = t; }
    else       { src = k; dst = kcodes; i = t - N; }
    const float4* p = (const float4*)(src + (size_t)i * DD);
    float4 a = p[0];
    float4 b = p[1];
    int code = (int)(a.x > 0.f)
             | ((int)(a.y > 0.f) << 1)
             | ((int)(a.z > 0.f) << 2)
             | ((int)(a.w > 0.f) << 3)
             | ((int)(b.x > 0.f) << 4)
             | ((int)(b.y > 0.f) << 5)
             | ((int)(b.z > 0.f) << 6)
             | ((int)(b.w > 0.f) << 7);
    dst[i] = code;
}

// ---------------------------------------------------------------------------
// Kernel 2: per (batch, code) bucket, collect the first <=64 matching key
// indices in ascending order via wave32 ballot compaction over LDS-resident
// key codes. Key codes are brought into LDS with the CDNA5 async path
// (global_load_async_to_lds_b128 + s_wait_asynccnt).
//
// Grid: BB*32 blocks of 256 threads (8 waves). Block = (batch b, code group
// of 8 codes); each wave owns exactly one code value.
// ---------------------------------------------------------------------------
__global__ void CandidateFinder_bucket(const int* __restrict__ kcodes,
                                       int* __restrict__ counts,
                                       int* __restrict__ table) {
    __shared__ int lds_k[LL];                      // 16 KB of this batch's key codes

    const int tid = threadIdx.x;
    const int b   = blockIdx.x >> 5;               // 32 blocks per batch
    const int cg  = blockIdx.x & 31;               // code group: 8 codes
    const int* kb = kcodes + (size_t)b * LL;

#if USE_ASYNC_LDS
    // Each thread issues 4 async B128 loads (16 ints): covers 4096 ints total.
    #pragma unroll
    for (int r = 0; r < 4; ++r) {
        int elt = (r * 256 + tid) * 4;             // first int of a 16-byte chunk
        unsigned lds_off = (unsigned)(uintptr_t)(&lds_k[elt]);   // LDS byte offset
        unsigned long long gaddr = (unsigned long long)(uintptr_t)(kb + elt);
        asm volatile("global_load_async_to_lds_b128 %0, %1, off"
                     :: "v"(lds_off), "v"(gaddr)
                     : "memory");
    }
    asm volatile("s_wait_asynccnt 0" ::: "memory");
#else
    #pragma unroll
    for (int r = 0; r < 16; ++r)
        lds_k[r * 256 + tid] = kb[r * 256 + tid];
#endif
    __syncthreads();

    const int lane = tid & 31;
    const int wave = tid >> 5;                     // 0..7
    const int code = cg * 8 + wave;                // 0..255
    const int slot = b * NCODES + code;
    int* __restrict__ tslot = table + (size_t)slot * KMAX;
    const unsigned below = (1u << lane) - 1u;      // lanes strictly below me (wave32)

    int m = 0;                                     // matches found so far (wave-uniform)
    for (int s = 0; s < LL / 32; ++s) {
        int kc = lds_k[s * 32 + lane];
        bool match = (kc == code);
        unsigned mask = __builtin_amdgcn_ballot_w32(match);
        int pos = m + __popc(mask & below);        // ascending order: s asc, lane asc
        if (match && pos < KMAX)
            tslot[pos] = s * 32 + lane;
        m += __popc(mask);
        if (m >= KMAX) break;                      // first 64 secured; rest irrelevant
    }
    if (lane == 0)
        counts[slot] = (m > KMAX) ? KMAX : m;
}

// ---------------------------------------------------------------------------
// Kernel 3: one thread per output element. out row = [-1 pads][matches asc],
// exactly matching jnp.sort over (-1-padded candidates). Coalesced int64 stores.
// ---------------------------------------------------------------------------
__global__ void CandidateFinder_emit(const int* __restrict__ qcodes,
                                     const int* __restrict__ counts,
                                     const int* __restrict__ table,
                                     long long* __restrict__ out) {
    int gid = blockIdx.x * blockDim.x + threadIdx.x;   // 0 .. B*L*KMAX-1
    int q   = gid >> 6;                                 // query index (b*L + i)
    int kk  = gid & (KMAX - 1);
    int b   = q >> 12;                                  // q / L (L = 4096)
    int code = qcodes[q];
    int slot = b * NCODES + code;
    int cnt  = counts[slot];
    int pad  = KMAX - cnt;
    long long v = -1;
    if (kk >= pad)
        v = (long long)table[(size_t)slot * KMAX + (kk - pad)];
    out[gid] = v;
}

// ---------------------------------------------------------------------------
// Launcher
// ---------------------------------------------------------------------------
extern "C" void kernel_launch(void* const* d_in, const int* in_sizes, int n_in,
                              void* d_out, int out_size, void* d_ws, size_t ws_size,
                              hipStream_t stream) {
    (void)in_sizes; (void)n_in; (void)out_size; (void)ws_size;

    const float* query_up = (const float*)d_in[0];   // [B, L, D] f32
    const float* key_up   = (const float*)d_in[1];   // [B, L, D] f32
    // d_in[2] = head_idx (unused by the math)

    long long* out = (long long*)d_out;              // [B, L, KMAX] int64

    // Workspace layout (ints): qcodes[B*L] | kcodes[B*L] | counts[B*256] | table[B*256*64]
    int* qcodes = (int*)d_ws;
    int* kcodes = qcodes + BB * LL;
    int* counts = kcodes + BB * LL;
    int* table  = counts + BB * NCODES;

    // 1) pack sign codes
    {
        int total = 2 * BB * LL;                     // 32768
        CandidateFinder_pack_codes<<<total / 256, 256, 0, stream>>>(
            query_up, key_up, qcodes, kcodes);
    }
    // 2) build per-(batch,code) first-64 index buckets
    {
        CandidateFinder_bucket<<<BB * 32, 256, 0, stream>>>(kcodes, counts, table);
    }
    // 3) expand buckets into the padded, sorted output
    {
        int total = BB * LL * KMAX;                  // 1,048,576
        CandidateFinder_emit<<<total / 256, 256, 0, stream>>>(qcodes, counts, table, out);
    }
}